// DGCNN_Seg_28621662061131
// MI455X (gfx1250) — compile-verified
//
#include <hip/hip_runtime.h>
#include <hip/hip_bf16.h>

typedef float v2f __attribute__((ext_vector_type(2)));
typedef float v4f __attribute__((ext_vector_type(4)));
typedef float v8f __attribute__((ext_vector_type(8)));

#define BN_SCALE  0.9999950000374997f
#define NEG_SLOPE 0.2f
#define NPTS      32768   // 8 * 4096
#define NPB       4096    // points per batch
#define KNN_K     20

// ---------------------------------------------------------------------------
// CDNA5 f32 WMMA: D(16x16,f32) = A(16x4,f32) * B(4x16,f32) + C
// A frag (v2f): lanes 0-15 hold M=lane, K=0/1; lanes 16-31 hold M=lane-16, K=2/3
// B frag (v2f): lanes 0-15 hold N=lane, K=0/1; lanes 16-31 hold N=lane-16, K=2/3
// C/D (v8f):    vgpr r: lanes 0-15 -> D[r][lane]; lanes 16-31 -> D[r+8][lane-16]
// ---------------------------------------------------------------------------
__device__ __forceinline__ v8f wmma_f32(v2f a, v2f b, v8f c) {
  return __builtin_amdgcn_wmma_f32_16x16x4_f32(
      /*neg_a=*/false, a, /*neg_b=*/false, b,
      /*c_mod=*/(short)0, c, /*reuse_a=*/false, /*reuse_b=*/false);
}

__device__ __forceinline__ float leaky(float v) {
  return v > 0.f ? v : NEG_SLOPE * v;
}

// Same-wave LDS ordering fence (tile buffers are wave-private; no block barrier
// needed: DS ops of one wave execute in order).
__device__ __forceinline__ void lds_fence_wave() {
  asm volatile("s_wait_dscnt 0" ::: "memory");
}

// ---------------------------------------------------------------------------
// Top-20 state kept in scratch (address taken); the rare replace-min path is a
// single out-of-line code copy: one dynamic-index scratch store + min rescan.
// Common path (candidate <= threshold) never touches this.
// ---------------------------------------------------------------------------
struct TopK {
  float tv[KNN_K];
  int   ti[KNN_K];
  float tmin;
  int   minpos;
};

__device__ __attribute__((noinline)) void topk_insert(TopK* s, float v, int j) {
  s->tv[s->minpos] = v;
  s->ti[s->minpos] = j;
  float mn = s->tv[0];
  int mp = 0;
  for (int q = 1; q < KNN_K; ++q) {
    float t = s->tv[q];
    if (t < mn) { mn = t; mp = q; }
  }
  s->tmin = mn;
  s->minpos = mp;
}

// ---------------------------------------------------------------------------
// Fold BN into per-channel scale/bias; build stacked [Wa ; Wb-Wa] weights.
// wc1 is padded to K=4 (last column zero).
// ---------------------------------------------------------------------------
__global__ __launch_bounds__(256) void prep_kernel(
    const float* __restrict__ w1, const float* __restrict__ g1, const float* __restrict__ b1,
    const float* __restrict__ w2, const float* __restrict__ g2, const float* __restrict__ b2,
    const float* __restrict__ w3, const float* __restrict__ g3, const float* __restrict__ b3,
    const float* __restrict__ w4, const float* __restrict__ g4, const float* __restrict__ b4,
    const float* __restrict__ gs1, const float* __restrict__ bs1, const float* __restrict__ bt1,
    const float* __restrict__ gs2, const float* __restrict__ bs2, const float* __restrict__ bt2,
    float* __restrict__ wc1, float* __restrict__ wc2, float* __restrict__ wc3,
    float* __restrict__ wc4, float* __restrict__ sb,
    float* __restrict__ sm1, float* __restrict__ bm1,
    float* __restrict__ sm2, float* __restrict__ bm2) {
  const int t = threadIdx.x;
  for (int i = t; i < 64 * 4; i += 256) {          // wc1: (128,4), K padded 3->4
    int o = i >> 2, c = i & 3;
    float va = (c < 3) ? w1[o * 6 + c] : 0.f;
    float vd = (c < 3) ? (w1[o * 6 + 3 + c] - w1[o * 6 + c]) : 0.f;
    wc1[o * 4 + c] = va;
    wc1[(64 + o) * 4 + c] = vd;
  }
  for (int i = t; i < 64 * 64; i += 256) {         // wc2..4: (128,64)
    int o = i >> 6, c = i & 63;
    wc2[o * 64 + c] = w2[o * 128 + c];
    wc2[(64 + o) * 64 + c] = w2[o * 128 + 64 + c] - w2[o * 128 + c];
    wc3[o * 64 + c] = w3[o * 128 + c];
    wc3[(64 + o) * 64 + c] = w3[o * 128 + 64 + c] - w3[o * 128 + c];
    wc4[o * 64 + c] = w4[o * 128 + c];
    wc4[(64 + o) * 64 + c] = w4[o * 128 + 64 + c] - w4[o * 128 + c];
  }
  if (t < 64) {                                    // per-layer BN scale / bias
    sb[0 * 128 + t] = BN_SCALE * g1[t]; sb[0 * 128 + 64 + t] = b1[t];
    sb[1 * 128 + t] = BN_SCALE * g2[t]; sb[1 * 128 + 64 + t] = b2[t];
    sb[2 * 128 + t] = BN_SCALE * g3[t]; sb[2 * 128 + 64 + t] = b3[t];
    sb[3 * 128 + t] = BN_SCALE * g4[t]; sb[3 * 128 + 64 + t] = b4[t];
  }
  {                                                // MLP: leaky(s*(A·W^T) + (bs*s + beta))
    float s1 = BN_SCALE * gs1[t]; sm1[t] = s1; bm1[t] = bs1[t] * s1 + bt1[t];
    float s2 = BN_SCALE * gs2[t]; sm2[t] = s2; bm2[t] = bs2[t] * s2 + bt2[t];
  }
}

// ---------------------------------------------------------------------------
// Pad x (NPTS,3) -> XP (NPTS,4) with zero 4th channel, and squared norms.
// ---------------------------------------------------------------------------
__global__ __launch_bounds__(256) void xp_xx_kernel(const float* __restrict__ x,
                                                    float* __restrict__ XP,
                                                    float* __restrict__ XX) {
  int p = blockIdx.x * blockDim.x + threadIdx.x;
  if (p >= NPTS) return;
  float a = x[p * 3 + 0], b = x[p * 3 + 1], c = x[p * 3 + 2];
  v4f o; o.x = a; o.y = b; o.z = c; o.w = 0.f;
  *(v4f*)(XP + p * 4) = o;
  XX[p] = a * a + b * b + c * c;
}

// ---------------------------------------------------------------------------
// Squared norms per point (C=64 path), vectorized b128 loads.
// ---------------------------------------------------------------------------
__global__ __launch_bounds__(256) void xx_kernel(const float* __restrict__ X, int ldx,
                                                 float* __restrict__ XX) {
  int p = blockIdx.x * blockDim.x + threadIdx.x;
  if (p >= NPTS) return;
  const v4f* r = (const v4f*)(X + (size_t)p * ldx);
  float s = 0.f;
#pragma unroll
  for (int c = 0; c < 16; ++c) {
    v4f v = r[c];
    s += v.x * v.x + v.y * v.y + v.z * v.z + v.w * v.w;
  }
  XX[p] = s;
}

// ---------------------------------------------------------------------------
// Fused KNN: WMMA distance tiles + split-lane top-20 selection.
// Each wave owns a 16-row block; lanes 0-15 scan cols 0-7 of their row,
// lanes 16-31 scan cols 8-15; the two partial top-20 lists merge at the end.
// Wave-private LDS; no block barriers; group-max pregate before inserts.
// ---------------------------------------------------------------------------
template <int C>
__global__ __launch_bounds__(128) void knn_kernel(const float* __restrict__ X, int ldx,
                                                  const float* __restrict__ XX,
                                                  int* __restrict__ IDX) {
  static_assert(C % 4 == 0, "C must be multiple of 4");
  constexpr int NCH = C / 4;
  __shared__ float tile[4][256];
  __shared__ float mergebuf[4][16 * 40];
  const int wave = threadIdx.x >> 5;
  const int lane = threadIdx.x & 31;
  const int i0 = (blockIdx.x * 4 + wave) * 16;   // global row of this wave's block
  const int jbase = (i0 >> 12) << 12;            // batch base row
  const int half = lane >> 4;
  const int l15 = lane & 15;
  const int koff = half * 2;
  const int colOff = half * 8;                   // selection column offset
  float* tileW = tile[wave];
  float* mergeW = mergebuf[wave];

  v2f afrag[NCH];
  {
    const float* arow = X + (size_t)(i0 + l15) * ldx + koff;
#pragma unroll
    for (int ch = 0; ch < NCH; ++ch) afrag[ch] = *(const v2f*)(arow + ch * 4);
  }
  float xxi[8];
#pragma unroll
  for (int r = 0; r < 8; ++r) xxi[r] = XX[i0 + r + half * 8];

  TopK st;
#pragma unroll
  for (int q = 0; q < KNN_K; ++q) { st.tv[q] = -3.0e38f; st.ti[q] = jbase; }
  st.tmin = -3.0e38f;
  st.minpos = 0;
  float tmin = -3.0e38f;                         // register mirror of st.tmin

  // strength-reduced running pointers (no per-tile 64-bit multiplies)
  const float* brow = X + (size_t)(jbase + l15) * ldx + koff;
  const float* xxp = XX + jbase + l15;

  for (int j0 = 0; j0 < NPB; j0 += 16, brow += (size_t)16 * ldx, xxp += 16) {
    v8f acc;
    if constexpr (NCH > 1) {                      // two independent WMMA chains
      v8f acc0 = {}, acc1 = {};
#pragma unroll
      for (int ch = 0; ch < NCH / 2; ++ch) {
        v2f bf0 = *(const v2f*)(brow + (2 * ch) * 4);
        v2f bf1 = *(const v2f*)(brow + (2 * ch + 1) * 4);
        acc0 = wmma_f32(afrag[2 * ch], bf0, acc0);
        acc1 = wmma_f32(afrag[2 * ch + 1], bf1, acc1);
      }
      acc = acc0 + acc1;
    } else {
      v8f acc0 = {};
      v2f bf = *(const v2f*)(brow);
      acc = wmma_f32(afrag[0], bf, acc0);
    }
    const float xxj = *xxp;
#pragma unroll
    for (int r = 0; r < 8; ++r) {
      float nd = 2.f * acc[r] - xxi[r] - xxj;     // -||xi - xj||^2
      tileW[(r + half * 8) * 16 + l15] = nd;
    }
    lds_fence_wave();
    // split-lane selection: this lane scans row l15, cols colOff..colOff+7
    {
      const v4f* rp = (const v4f*)(tileW + l15 * 16 + colOff);
      v4f c0 = rp[0];
      v4f c1 = rp[1];
      float gm = fmaxf(fmaxf(fmaxf(c0.x, c0.y), fmaxf(c0.z, c0.w)),
                       fmaxf(fmaxf(c1.x, c1.y), fmaxf(c1.z, c1.w)));
      if (gm > tmin) {                            // usually false after warmup
        const int jb = jbase + j0 + colOff;
        if (c0.x > tmin) { topk_insert(&st, c0.x, jb + 0); tmin = st.tmin; }
        if (c0.y > tmin) { topk_insert(&st, c0.y, jb + 1); tmin = st.tmin; }
        if (c0.z > tmin) { topk_insert(&st, c0.z, jb + 2); tmin = st.tmin; }
        if (c0.w > tmin) { topk_insert(&st, c0.w, jb + 3); tmin = st.tmin; }
        if (c1.x > tmin) { topk_insert(&st, c1.x, jb + 4); tmin = st.tmin; }
        if (c1.y > tmin) { topk_insert(&st, c1.y, jb + 5); tmin = st.tmin; }
        if (c1.z > tmin) { topk_insert(&st, c1.z, jb + 6); tmin = st.tmin; }
        if (c1.w > tmin) { topk_insert(&st, c1.w, jb + 7); tmin = st.tmin; }
      }
    }
    lds_fence_wave();   // reads done before next iteration overwrites the tile
  }

  // merge the two half-lane lists for each row
  if (lane >= 16) {
    for (int q = 0; q < KNN_K; ++q) {
      mergeW[l15 * 40 + q] = st.tv[q];
      mergeW[l15 * 40 + 20 + q] = __int_as_float(st.ti[q]);
    }
  }
  lds_fence_wave();
  if (lane < 16) {
    for (int q = 0; q < KNN_K; ++q) {
      float v = mergeW[l15 * 40 + q];
      int j = __float_as_int(mergeW[l15 * 40 + 20 + q]);
      if (v > tmin) { topk_insert(&st, v, j); tmin = st.tmin; }
    }
    int p = i0 + l15;
    for (int q = 0; q < KNN_K; ++q) IDX[p * KNN_K + q] = st.ti[q];
  }
}

// ---------------------------------------------------------------------------
// WMMA f32 GEMM, 32x64 output per wave (2x4 tiles, 8 accumulators):
// Out = act(scale .* (A[M x K] * W[Nout x K]^T) + bias)
// Requires: M % 32 == 0, Nout % 64 == 0, K % 4 == 0 (compile-time K).
// ---------------------------------------------------------------------------
template <int K>
__global__ __launch_bounds__(128) void gemm2x4_kernel(
    const float* __restrict__ A, int lda,
    const float* __restrict__ W,
    float* __restrict__ Out, int ldo,
    const float* __restrict__ scale, const float* __restrict__ bias,
    int act, int Mt2, int Nt4) {
  const int wave = threadIdx.x >> 5;
  const int lane = threadIdx.x & 31;
  const int tid = blockIdx.x * 4 + wave;          // wave-uniform tile id
  if (tid >= Mt2 * Nt4) return;
  const int tn = tid % Nt4, tm = tid / Nt4;
  const int half = lane >> 4, l15 = lane & 15, koff = half * 2;
  const int row0 = tm * 32 + l15;
  const int col0 = tn * 64 + l15;
  const float* a0 = A + (size_t)row0 * lda + koff;
  const float* a1 = a0 + (size_t)16 * lda;
  const float* wp = W + (size_t)col0 * K + koff;

  v8f acc[2][4];
#pragma unroll
  for (int i = 0; i < 2; ++i)
#pragma unroll
    for (int j = 0; j < 4; ++j) acc[i][j] = {};

#pragma unroll 4
  for (int c0 = 0; c0 < K; c0 += 4) {
    v2f aA = *(const v2f*)(a0 + c0);
    v2f aB = *(const v2f*)(a1 + c0);
    v2f b[4];
#pragma unroll
    for (int j = 0; j < 4; ++j) b[j] = *(const v2f*)(wp + (size_t)(16 * j) * K + c0);
#pragma unroll
    for (int j = 0; j < 4; ++j) {
      acc[0][j] = wmma_f32(aA, b[j], acc[0][j]);
      acc[1][j] = wmma_f32(aB, b[j], acc[1][j]);
    }
  }

  float sc[4], bi[4];
#pragma unroll
  for (int j = 0; j < 4; ++j) {
    sc[j] = scale ? scale[col0 + 16 * j] : 1.f;
    bi[j] = bias ? bias[col0 + 16 * j] : 0.f;
  }
#pragma unroll
  for (int r = 0; r < 8; ++r) {
    const int orow0 = tm * 32 + r + half * 8;
    const int orow1 = orow0 + 16;
#pragma unroll
    for (int j = 0; j < 4; ++j) {
      float v0 = acc[0][j][r] * sc[j] + bi[j];
      float v1 = acc[1][j][r] * sc[j] + bi[j];
      if (act) { v0 = leaky(v0); v1 = leaky(v1); }
      Out[(size_t)orow0 * ldo + col0 + 16 * j] = v0;
      Out[(size_t)orow1 * ldo + col0 + 16 * j] = v1;
    }
  }
}

// ---------------------------------------------------------------------------
// Narrow-output GEMM (Nout < 16, e.g. logits Nout=2): one 16x16 tile per wave,
// weight row index clamped + fragment zeroed by select (no branchy loads).
// ---------------------------------------------------------------------------
template <int K>
__global__ __launch_bounds__(128) void gemm_tail_kernel(
    const float* __restrict__ A, int lda,
    const float* __restrict__ W, int Nout,
    float* __restrict__ Out, int ldo,
    const float* __restrict__ bias, int Mt) {
  const int wave = threadIdx.x >> 5;
  const int lane = threadIdx.x & 31;
  const int tm = blockIdx.x * 4 + wave;
  if (tm >= Mt) return;
  const int half = lane >> 4, l15 = lane & 15, koff = half * 2;
  const int row = tm * 16 + l15;
  const bool colok = l15 < Nout;
  const int wrow = colok ? l15 : (Nout - 1);      // clamped: loads always valid
  const float* a0 = A + (size_t)row * lda + koff;
  const float* w0 = W + (size_t)wrow * K + koff;

  v8f acc = {};
#pragma unroll 8
  for (int c0 = 0; c0 < K; c0 += 4) {
    v2f a = *(const v2f*)(a0 + c0);
    v2f b = *(const v2f*)(w0 + c0);
    b.x = colok ? b.x : 0.f;                      // v_cndmask, no EXEC churn
    b.y = colok ? b.y : 0.f;
    acc = wmma_f32(a, b, acc);
  }
  if (colok) {
    const float bi = bias ? bias[l15] : 0.f;
#pragma unroll
    for (int r = 0; r < 8; ++r) {
      const int orow = tm * 16 + r + half * 8;
      Out[(size_t)orow * ldo + l15] = acc[r] + bi;
    }
  }
}

// ---------------------------------------------------------------------------
// Sign-aware neighbor reduce + fused BN/LeakyReLU:
// out[p,o] = leaky(s_o * (opt_k P[j_k,o] + Q[p,o]) + b_o),  opt = max if s>=0 else min
// PQ rows: [P(64) | Q(64)]. One thread per (p, o).
// ---------------------------------------------------------------------------
__global__ __launch_bounds__(256) void edge_reduce_kernel(
    const float* __restrict__ PQ, const int* __restrict__ IDX,
    const float* __restrict__ S, const float* __restrict__ Bc,
    float* __restrict__ H, int ldo) {
  int t = blockIdx.x * 256 + threadIdx.x;
  int o = t & 63;
  int p = t >> 6;
  float s = S[o], bb = Bc[o];
  float q = PQ[(size_t)p * 128 + 64 + o];
  bool useMax = s >= 0.f;
  float red = useMax ? -3.0e38f : 3.0e38f;
  const int* ip = IDX + p * KNN_K;
#pragma unroll 5
  for (int k = 0; k < KNN_K; ++k) {
    int j = ip[k];                                 // global row index
    float v = PQ[(size_t)j * 128 + o];
    red = useMax ? fmaxf(red, v) : fminf(red, v);
  }
  float z = s * (red + q) + bb;
  H[(size_t)p * ldo + o] = leaky(z);
}

// ---------------------------------------------------------------------------
extern "C" void kernel_launch(void* const* d_in, const int* in_sizes, int n_in,
                              void* d_out, int out_size, void* d_ws, size_t ws_size,
                              hipStream_t stream) {
  (void)in_sizes; (void)n_in; (void)out_size; (void)ws_size;
  const float* x   = (const float*)d_in[0];
  const float* w1  = (const float*)d_in[1];
  const float* g1  = (const float*)d_in[2];
  const float* b1  = (const float*)d_in[3];
  const float* w2  = (const float*)d_in[4];
  const float* g2  = (const float*)d_in[5];
  const float* b2  = (const float*)d_in[6];
  const float* w3  = (const float*)d_in[7];
  const float* g3  = (const float*)d_in[8];
  const float* b3  = (const float*)d_in[9];
  const float* w4  = (const float*)d_in[10];
  const float* g4  = (const float*)d_in[11];
  const float* b4  = (const float*)d_in[12];
  const float* ws1 = (const float*)d_in[13];
  const float* bs1 = (const float*)d_in[14];
  const float* gs1 = (const float*)d_in[15];
  const float* bt1 = (const float*)d_in[16];
  const float* ws2 = (const float*)d_in[17];
  const float* bs2 = (const float*)d_in[18];
  const float* gs2 = (const float*)d_in[19];
  const float* bt2 = (const float*)d_in[20];
  const float* ws3 = (const float*)d_in[21];
  const float* bs3 = (const float*)d_in[22];
  float* out = (float*)d_out;

  float* ws = (float*)d_ws;
  float* H   = ws;                       // (32768, 256)
  float* Y1  = H  + (size_t)NPTS * 256;  // (32768, 256)
  float* PQ  = Y1 + (size_t)NPTS * 256;  // (32768, 128)
  float* XX  = PQ + (size_t)NPTS * 128;  // (32768,)
  float* XP  = XX + NPTS;                // (32768, 4) padded x
  int*   IDX = (int*)(XP + (size_t)NPTS * 4);   // (32768, 20)
  float* WC1 = (float*)(IDX + (size_t)NPTS * KNN_K);
  float* WC2 = WC1 + 128 * 4;
  float* WC3 = WC2 + 128 * 64;
  float* WC4 = WC3 + 128 * 64;
  float* SB  = WC4 + 128 * 64;           // 4 layers x [s(64) | b(64)]
  float* SM1 = SB + 512;
  float* BM1 = SM1 + 256;
  float* SM2 = BM1 + 256;
  float* BM2 = SM2 + 256;
  float* Y2  = H;                        // reuse H for second MLP output

  prep_kernel<<<1, 256, 0, stream>>>(w1, g1, b1, w2, g2, b2, w3, g3, b3, w4, g4, b4,
                                     gs1, bs1, bt1, gs2, bs2, bt2,
                                     WC1, WC2, WC3, WC4, SB, SM1, BM1, SM2, BM2);

  const int Mt2 = NPTS / 32;   // 1024 row tiles (32-wide)
  const int Mt  = NPTS / 16;   // 2048 row tiles (16-wide)

  // ---- edge conv 1 (C=3 padded to 4) ----
  xp_xx_kernel<<<128, 256, 0, stream>>>(x, XP, XX);
  knn_kernel<4><<<512, 128, 0, stream>>>(XP, 4, XX, IDX);
  gemm2x4_kernel<4><<<(Mt2 * 2 + 3) / 4, 128, 0, stream>>>(XP, 4, WC1, PQ, 128,
                                                           nullptr, nullptr, 0, Mt2, 2);
  edge_reduce_kernel<<<NPTS * 64 / 256, 256, 0, stream>>>(PQ, IDX, SB + 0, SB + 64, H + 0, 256);

  // ---- edge conv 2 ----
  xx_kernel<<<128, 256, 0, stream>>>(H + 0, 256, XX);
  knn_kernel<64><<<512, 128, 0, stream>>>(H + 0, 256, XX, IDX);
  gemm2x4_kernel<64><<<(Mt2 * 2 + 3) / 4, 128, 0, stream>>>(H + 0, 256, WC2, PQ, 128,
                                                            nullptr, nullptr, 0, Mt2, 2);
  edge_reduce_kernel<<<NPTS * 64 / 256, 256, 0, stream>>>(PQ, IDX, SB + 128, SB + 192, H + 64, 256);

  // ---- edge conv 3 ----
  xx_kernel<<<128, 256, 0, stream>>>(H + 64, 256, XX);
  knn_kernel<64><<<512, 128, 0, stream>>>(H + 64, 256, XX, IDX);
  gemm2x4_kernel<64><<<(Mt2 * 2 + 3) / 4, 128, 0, stream>>>(H + 64, 256, WC3, PQ, 128,
                                                            nullptr, nullptr, 0, Mt2, 2);
  edge_reduce_kernel<<<NPTS * 64 / 256, 256, 0, stream>>>(PQ, IDX, SB + 256, SB + 320, H + 128, 256);

  // ---- edge conv 4 ----
  xx_kernel<<<128, 256, 0, stream>>>(H + 128, 256, XX);
  knn_kernel<64><<<512, 128, 0, stream>>>(H + 128, 256, XX, IDX);
  gemm2x4_kernel<64><<<(Mt2 * 2 + 3) / 4, 128, 0, stream>>>(H + 128, 256, WC4, PQ, 128,
                                                            nullptr, nullptr, 0, Mt2, 2);
  edge_reduce_kernel<<<NPTS * 64 / 256, 256, 0, stream>>>(PQ, IDX, SB + 384, SB + 448, H + 192, 256);

  // ---- MLP head ----
  gemm2x4_kernel<256><<<(Mt2 * 4 + 3) / 4, 128, 0, stream>>>(H, 256, ws1, Y1, 256,
                                                             SM1, BM1, 1, Mt2, 4);
  gemm2x4_kernel<256><<<(Mt2 * 4 + 3) / 4, 128, 0, stream>>>(Y1, 256, ws2, Y2, 256,
                                                             SM2, BM2, 1, Mt2, 4);
  gemm_tail_kernel<256><<<(Mt + 3) / 4, 128, 0, stream>>>(Y2, 256, ws3, 2, out, 2,
                                                          bs3, Mt);
}